// ProposalLayer_78743930404873
// MI455X (gfx1250) — compile-verified
//
#include <hip/hip_runtime.h>
#include <hip/hip_bf16.h>
#include <cstdint>
#include <cstddef>

// ---------------------------------------------------------------------------
// ProposalLayer for MI455X (gfx1250).
// Latency/bandwidth-bound layer (no matrix ops -> no WMMA role).
// CDNA5-specific paths used:
//   * global_load_async_to_lds_b128 + s_wait_asynccnt  (bitonic chunk staging
//     and NMS column-tile staging straight into LDS, tracked on ASYNCcnt)
//   * global_prefetch_b8 via __builtin_prefetch        (serial NMS scan)
//   * wave32-sized greedy scan (one wave owns the 192-word suppress bitmap)
// Bitonic sort fused: all intra-512-chunk stages run inside LDS kernels,
// cutting kernel launches from 158 to 50 on this launch-latency-bound layer.
// ---------------------------------------------------------------------------

#define ANUM   9
#define HH     100
#define WW     100
#define NA     (HH * WW * ANUM)   /* 90000 anchors */
#define NPAD   131072             /* 2^17, bitonic sort size */
#define CHUNK  512                /* LDS-resident bitonic chunk */
#define PRE    6000               /* PRE_NMS */
#define POST   300                /* POST_NMS */
#define THRESH 0.7f
#define WORDS  192                /* 24 col tiles * 8 words = covers 6144 cols */

// Async DMA of 16 bytes from global straight into LDS (per lane).
__device__ __forceinline__ void async_copy_b128(void* lds_dst, const void* gsrc)
{
    unsigned ldsaddr = (unsigned)(uintptr_t)lds_dst;   // low 32 bits of generic
                                                       // pointer == LDS offset
    unsigned long long gaddr = (unsigned long long)(uintptr_t)gsrc;
    asm volatile("global_load_async_to_lds_b128 %0, %1, off"
                 :: "v"(ldsaddr), "v"(gaddr) : "memory");
}
__device__ __forceinline__ void async_wait0()
{
    asm volatile("s_wait_asynccnt 0" ::: "memory");
}

// ---------------------------------------------------------------------------
// Kernel 1: box transform + clip, and build 64-bit sort keys.
// key = (scoreBits << 32) | (0xFFFFFFFF - idx)  -> descending sort gives
// descending score with ascending-index tie-break == stable argsort(-scores).
// ---------------------------------------------------------------------------
__global__ void k_prop_init(const float* __restrict__ anchors,
                            const float* __restrict__ cls,
                            const float* __restrict__ bbox,
                            const float* __restrict__ iminfo,
                            float4* __restrict__ boxes,
                            unsigned long long* __restrict__ keys)
{
    int i = blockIdx.x * blockDim.x + threadIdx.x;
    if (i >= NPAD) return;
    if (i >= NA) { keys[i] = 0ull; return; }   // pad: score 0.0, worst index

    int c = i / ANUM;
    int a = i - c * ANUM;

    float score = cls[c * (2 * ANUM) + ANUM + a];
    unsigned sb = __float_as_uint(score);      // scores >= 0 -> bits monotonic
    keys[i] = ((unsigned long long)sb << 32) |
              (unsigned long long)(0xFFFFFFFFu - (unsigned)i);

    float ax1 = anchors[4 * i + 0], ay1 = anchors[4 * i + 1];
    float ax2 = anchors[4 * i + 2], ay2 = anchors[4 * i + 3];
    float w  = ax2 - ax1 + 1.0f;
    float h  = ay2 - ay1 + 1.0f;
    float cx = ax1 + 0.5f * w;
    float cy = ay1 + 0.5f * h;

    float dx = bbox[4 * i + 0], dy = bbox[4 * i + 1];
    float dw = bbox[4 * i + 2], dh = bbox[4 * i + 3];

    float pcx = dx * w + cx;
    float pcy = dy * h + cy;
    float pw  = expf(dw) * w;
    float ph  = expf(dh) * h;

    float x1 = pcx - 0.5f * pw;
    float y1 = pcy - 0.5f * ph;
    float x2 = pcx + 0.5f * pw;
    float y2 = pcy + 0.5f * ph;

    float imh = iminfo[0], imw = iminfo[1];
    x1 = fminf(fmaxf(x1, 0.0f), imw - 1.0f);
    y1 = fminf(fmaxf(y1, 0.0f), imh - 1.0f);
    x2 = fminf(fmaxf(x2, 0.0f), imw - 1.0f);
    y2 = fminf(fmaxf(y2, 0.0f), imh - 1.0f);

    boxes[i] = make_float4(x1, y1, x2, y2);
}

// ---------------------------------------------------------------------------
// Bitonic: global compare-exchange step (only used while j >= CHUNK).
// ---------------------------------------------------------------------------
__global__ void k_bitonic_global(unsigned long long* __restrict__ keys,
                                 int j, int k)
{
    int i = blockIdx.x * blockDim.x + threadIdx.x;
    int ixj = i ^ j;
    if (ixj <= i) return;
    unsigned long long a = keys[i];
    unsigned long long b = keys[ixj];
    bool desc = ((i & k) == 0);
    if (desc ? (a < b) : (a > b)) {
        keys[i]   = b;
        keys[ixj] = a;
    }
}

// ---------------------------------------------------------------------------
// Bitonic: fused LDS kernel covering ALL stages k=2..CHUNK (partners stay
// inside a 512-key chunk). Chunk staged into LDS by async DMA (16B/lane).
// Pair mapping: thread t handles i = ((t & ~(j-1)) << 1) | (t & (j-1)),
// partner i+j -> exactly one pair per thread per step.
// ---------------------------------------------------------------------------
__global__ void k_bitonic_local_full(unsigned long long* __restrict__ keys)
{
    __shared__ unsigned long long lk[CHUNK];            // 4 KB
    const int t    = threadIdx.x;                       // 256 threads
    const int base = blockIdx.x * CHUNK;

    async_copy_b128(&lk[2 * t], keys + base + 2 * t);   // two keys per lane
    async_wait0();
    __syncthreads();

    for (int k = 2; k <= CHUNK; k <<= 1) {
        for (int j = k >> 1; j > 0; j >>= 1) {
            const int  i    = ((t & ~(j - 1)) << 1) | (t & (j - 1));
            const int  p    = i + j;
            const bool desc = (((base + i) & k) == 0);
            unsigned long long a = lk[i], b = lk[p];
            if (desc ? (a < b) : (a > b)) { lk[i] = b; lk[p] = a; }
            __syncthreads();
        }
    }
    keys[base + t]       = lk[t];
    keys[base + t + 256] = lk[t + 256];
}

// ---------------------------------------------------------------------------
// Bitonic: fused LDS merge for one k >= 1024, covering j=256..1.
// Direction is uniform per chunk (k > CHUNK).
// ---------------------------------------------------------------------------
__global__ void k_bitonic_local_merge(unsigned long long* __restrict__ keys,
                                      int k)
{
    __shared__ unsigned long long lk[CHUNK];
    const int t    = threadIdx.x;
    const int base = blockIdx.x * CHUNK;

    async_copy_b128(&lk[2 * t], keys + base + 2 * t);
    async_wait0();
    __syncthreads();

    const bool desc = ((base & k) == 0);
    for (int j = CHUNK >> 1; j > 0; j >>= 1) {
        const int i = ((t & ~(j - 1)) << 1) | (t & (j - 1));
        const int p = i + j;
        unsigned long long a = lk[i], b = lk[p];
        if (desc ? (a < b) : (a > b)) { lk[i] = b; lk[p] = a; }
        __syncthreads();
    }
    keys[base + t]       = lk[t];
    keys[base + t + 256] = lk[t + 256];
}

// ---------------------------------------------------------------------------
// Gather the top-6000 (sorted) boxes / scores / trans rows.
// ---------------------------------------------------------------------------
__global__ void k_gather(const unsigned long long* __restrict__ keys,
                         const float4* __restrict__ boxes,
                         const float* __restrict__ trans_in,
                         float4* __restrict__ sboxes,
                         float* __restrict__ sscores,
                         float* __restrict__ strans)
{
    int r = blockIdx.x * blockDim.x + threadIdx.x;
    if (r >= PRE) return;
    unsigned long long key = keys[r];
    unsigned idx = 0xFFFFFFFFu - (unsigned)(key & 0xFFFFFFFFull);
    sboxes[r]  = boxes[idx];
    sscores[r] = __uint_as_float((unsigned)(key >> 32));
#pragma unroll
    for (int t = 0; t < 6; ++t)
        strans[r * 6 + t] = trans_in[(size_t)idx * 6 + t];
}

// ---------------------------------------------------------------------------
// NMS suppression bit-matrix. Each block: 256 rows x 256 cols.
// Column boxes staged into LDS via async DMA. IoU test is division-free:
// inter > t*(a_i + a_j - inter)  (denominator positive after clipping).
// ---------------------------------------------------------------------------
__global__ void k_nms_mask(const float4* __restrict__ sboxes,
                           unsigned* __restrict__ mask)
{
    __shared__ float4 cb[256];          // 4 KB column-box tile

    const int tid     = threadIdx.x;
    const int colBase = blockIdx.x * 256;
    const int r       = blockIdx.y * 256 + tid;

    int jc = colBase + tid;
    if (jc >= PRE) jc = 0;                          // clamp pad lanes
    async_copy_b128(&cb[tid], sboxes + jc);
    async_wait0();
    __syncthreads();

    if (r >= PRE) return;

    const float4 rb = sboxes[r];
    const float  ra = (rb.z - rb.x + 1.0f) * (rb.w - rb.y + 1.0f);

#pragma unroll
    for (int w = 0; w < 8; ++w) {
        unsigned bits = 0u;
#pragma unroll
        for (int b = 0; b < 32; ++b) {
            const int jj = w * 32 + b;
            const int jg = colBase + jj;
            const float4 c = cb[jj];
            float xx1 = fmaxf(rb.x, c.x);
            float yy1 = fmaxf(rb.y, c.y);
            float xx2 = fminf(rb.z, c.z);
            float yy2 = fminf(rb.w, c.w);
            float iw  = fmaxf(xx2 - xx1 + 1.0f, 0.0f);
            float ih  = fmaxf(yy2 - yy1 + 1.0f, 0.0f);
            float inter = iw * ih;
            float ca = (c.z - c.x + 1.0f) * (c.w - c.y + 1.0f);
            bool sup = (jg > r) & (jg < PRE) &
                       (inter > THRESH * (ra + ca - inter));
            bits |= ((unsigned)sup) << b;
        }
        mask[(size_t)r * WORDS + (colBase >> 5) + w] = bits;
    }
}

// ---------------------------------------------------------------------------
// Greedy NMS scan: one wave32. 192-word suppression bitmap lives in LDS
// (6 words per lane). Prefetch the next bitmask row while OR-ing this one.
// Produces keep_idx[300] with fill_value=0 semantics (matches jnp.where).
// ---------------------------------------------------------------------------
__global__ void k_nms_scan(const unsigned* __restrict__ mask,
                           int* __restrict__ keep_idx)
{
    __shared__ unsigned sup[WORDS];
    const int lane = threadIdx.x;          // blockDim.x == 32 (one wave)

    for (int w = lane; w < WORDS; w += 32) sup[w] = 0u;
    __syncthreads();

    int count = 0;                          // uniform across the wave
    for (int i = 0; i < PRE; ++i) {
        if (i + 1 < PRE)
            __builtin_prefetch(mask + (size_t)(i + 1) * WORDS, 0, 0);

        const bool suppressed = (sup[i >> 5] >> (i & 31)) & 1u;
        if (!suppressed) {
            if (count < POST && lane == 0) keep_idx[count] = i;
            ++count;
            for (int w = lane; w < WORDS; w += 32)
                sup[w] |= mask[(size_t)i * WORDS + w];
        }
        __syncthreads();
    }

    const int start = (count < POST) ? count : POST;
    for (int r = start + lane; r < POST; r += 32) keep_idx[r] = 0;
}

// ---------------------------------------------------------------------------
// Emit the concatenated outputs: blob[300*5] | scores[300] | trans[300*6].
// ---------------------------------------------------------------------------
__global__ void k_output(const int* __restrict__ keep_idx,
                         const float4* __restrict__ sboxes,
                         const float* __restrict__ sscores,
                         const float* __restrict__ strans,
                         float* __restrict__ out)
{
    int r = blockIdx.x * blockDim.x + threadIdx.x;
    if (r >= POST) return;
    int i = keep_idx[r];
    float4 b = sboxes[i];
    out[r * 5 + 0] = 0.0f;                 // batch index
    out[r * 5 + 1] = b.x;
    out[r * 5 + 2] = b.y;
    out[r * 5 + 3] = b.z;
    out[r * 5 + 4] = b.w;
    out[POST * 5 + r] = sscores[i];
#pragma unroll
    for (int t = 0; t < 6; ++t)
        out[POST * 5 + POST + r * 6 + t] = strans[i * 6 + t];
}

// ---------------------------------------------------------------------------
extern "C" void kernel_launch(void* const* d_in, const int* in_sizes, int n_in,
                              void* d_out, int out_size, void* d_ws, size_t ws_size,
                              hipStream_t stream)
{
    (void)in_sizes; (void)n_in; (void)out_size; (void)ws_size;

    const float* anchors = (const float*)d_in[0];   // [90000,4]
    const float* cls     = (const float*)d_in[1];   // [1,100,100,18]
    const float* bbox    = (const float*)d_in[2];   // [1,100,100,36]
    const float* trans   = (const float*)d_in[3];   // [1,100,100,54]
    const float* iminfo  = (const float*)d_in[4];   // [3]
    float*       out     = (float*)d_out;           // 3600 floats

    // -------- workspace layout (all 16-byte aligned) --------
    char* ws = (char*)d_ws;
    size_t off = 0;
    float4* boxes = (float4*)(ws + off);                 off += (size_t)NA * 16;       // 1,440,000
    unsigned long long* keys = (unsigned long long*)(ws + off); off += (size_t)NPAD * 8; // 1,048,576
    float4* sboxes = (float4*)(ws + off);                off += (size_t)PRE * 16;      //    96,000
    float*  sscores = (float*)(ws + off);                off += (size_t)PRE * 4;       //    24,000
    float*  strans  = (float*)(ws + off);                off += (size_t)PRE * 6 * 4;   //   144,000
    unsigned* mask  = (unsigned*)(ws + off);             off += (size_t)PRE * WORDS * 4;// 4,608,000
    int* keep_idx   = (int*)(ws + off);                  off += (size_t)POST * 4;
    // total ~7.4 MB

    // 1) transform + clip + sort keys (padded to 2^17)
    k_prop_init<<<NPAD / 256, 256, 0, stream>>>(anchors, cls, bbox, iminfo,
                                                boxes, keys);

    // 2) bitonic sort, descending, stable tie-break via key low bits.
    //    All intra-chunk (512-key) stages fused into LDS kernels.
    k_bitonic_local_full<<<NPAD / CHUNK, 256, 0, stream>>>(keys);   // k=2..512
    for (int k = CHUNK * 2; k <= NPAD; k <<= 1) {
        for (int j = k >> 1; j >= CHUNK; j >>= 1)
            k_bitonic_global<<<NPAD / 256, 256, 0, stream>>>(keys, j, k);
        k_bitonic_local_merge<<<NPAD / CHUNK, 256, 0, stream>>>(keys, k);
    }

    // 3) gather sorted top-6000
    k_gather<<<(PRE + 255) / 256, 256, 0, stream>>>(keys, boxes, trans,
                                                    sboxes, sscores, strans);

    // 4) pairwise IoU suppression bit-matrix (async-LDS staged tiles)
    dim3 mg((PRE + 255) / 256, (PRE + 255) / 256);       // 24 x 24
    k_nms_mask<<<mg, 256, 0, stream>>>(sboxes, mask);

    // 5) greedy scan on one wave32
    k_nms_scan<<<1, 32, 0, stream>>>(mask, keep_idx);

    // 6) outputs
    k_output<<<(POST + 255) / 256, 256, 0, stream>>>(keep_idx, sboxes, sscores,
                                                     strans, out);
}